// GTFB_64158221468246
// MI455X (gfx1250) — compile-verified
//
#include <hip/hip_runtime.h>

// Problem constants (from reference setup_inputs)
static constexpr int kB = 8;       // batch
static constexpr int kT = 16384;   // samples
static constexpr int kF = 256;     // filters
static constexpr int kL = 256;     // chunk length
static constexpr int kC = kT / kL; // 64 chunks
static constexpr int kN = kC * kB; // 512 (chunk,batch) columns

typedef __attribute__((ext_vector_type(2))) float v2f;
typedef __attribute__((ext_vector_type(8))) float v8f;
typedef __attribute__((ext_vector_type(4))) unsigned int v4u;
typedef __attribute__((ext_vector_type(8))) int v8i;
typedef __attribute__((ext_vector_type(4))) int v4i;

// ---------------------------------------------------------------------------
// Tensor Data Mover: 2D tile load Global -> LDS (6-arg builtin, clang-23 lane).
// Descriptor bit layout per CDNA5 ISA (D# groups 0/1).
// dims/strides/tiles are in element (data_size=4B) units.
// ---------------------------------------------------------------------------
__device__ static inline void tdm_load_2d_f32(unsigned int lds_byte_addr,
                                              const void* gaddr,
                                              unsigned int dim0, unsigned int dim1,
                                              unsigned int stride0,
                                              unsigned int tile0, unsigned int tile1) {
  const unsigned long long ga = (unsigned long long)(uintptr_t)gaddr;
  v4u g0;
  g0[0] = 1u;                                            // count=1, is_restore=0, no gather
  g0[1] = lds_byte_addr;                                 // lds_addr (bytes)
  g0[2] = (unsigned int)(ga & 0xffffffffu);              // global_addr[31:0]
  g0[3] = (unsigned int)((ga >> 32) & 0x01ffffffu)       // global_addr[56:32]
        | (2u << 30);                                    // type=2 ("image")
  v8i g1;
  g1[0] = (int)(2u << 16);                               // wg_mask=0, data_size=2 (4B)
  g1[1] = (int)((dim0 & 0xffffu) << 16);                 // tensor_dim0[15:0] @ bit48
  g1[2] = (int)((dim0 >> 16) | ((dim1 & 0xffffu) << 16));// tensor_dim0 hi | tensor_dim1 lo
  g1[3] = (int)((dim1 >> 16) | ((tile0 & 0xffffu) << 16));// tensor_dim1 hi | tile_dim0
  g1[4] = (int)(tile1 & 0xffffu);                        // tile_dim1 (tile_dim2=0)
  g1[5] = (int)stride0;                                  // tensor_dim0_stride[31:0]
  g1[6] = 0;                                             // stride0 hi | stride1 lo
  g1[7] = 0;                                             // stride1 hi
  const v4i gz4 = {0, 0, 0, 0};                          // 2D: groups 2/3 unused
  const v8i gz8 = {0, 0, 0, 0, 0, 0, 0, 0};              // extra group (clang-23 form)
  __builtin_amdgcn_tensor_load_to_lds(g0, g1, gz4, gz4, gz8, 0);
}

// ---------------------------------------------------------------------------
// K0: per filter, generate impulse response h[0..L-1] (complex) and the four
// homogeneous responses G_j[0..L-1] (unit initial state on y_{-1-j}).
// ---------------------------------------------------------------------------
__global__ void gtfb_coeffs_k(const float* __restrict__ nf,
                              const float* __restrict__ a1,
                              const float* __restrict__ a2,
                              const float* __restrict__ a3,
                              const float* __restrict__ a4,
                              float* __restrict__ Hre, float* __restrict__ Him,
                              float* __restrict__ HreT,
                              float* __restrict__ Gre, float* __restrict__ Gim) {
  const int f = blockIdx.x;
  const int j = threadIdx.x;          // 0..3 -> G_j, 4 -> h
  if (j >= 5) return;
  const float a1r = a1[2 * f], a1i = a1[2 * f + 1];
  const float a2r = a2[2 * f], a2i = a2[2 * f + 1];
  const float a3r = a3[2 * f], a3i = a3[2 * f + 1];
  const float a4r = a4[2 * f], a4i = a4[2 * f + 1];
  float y1r = 0.f, y1i = 0.f, y2r = 0.f, y2i = 0.f;
  float y3r = 0.f, y3i = 0.f, y4r = 0.f, y4i = 0.f;
  if (j == 0) y1r = 1.f;              // y_{-1} = 1
  else if (j == 1) y2r = 1.f;         // y_{-2} = 1
  else if (j == 2) y3r = 1.f;
  else if (j == 3) y4r = 1.f;
  const float x0 = (j == 4) ? nf[f] : 0.f;
  for (int n = 0; n < kL; ++n) {
    const float xr = (n == 0) ? x0 : 0.f;
    const float yr = xr + a1r * y1r - a1i * y1i + a2r * y2r - a2i * y2i
                        + a3r * y3r - a3i * y3i + a4r * y4r - a4i * y4i;
    const float yi =      a1r * y1i + a1i * y1r + a2r * y2i + a2i * y2r
                        + a3r * y3i + a3i * y3r + a4r * y4i + a4i * y4r;
    if (j == 4) {
      Hre[f * kL + n] = yr;
      Him[f * kL + n] = yi;
      HreT[n * kF + f] = yr;          // transposed slab for the WMMA B operand
    } else {
      Gre[((size_t)f * kL + n) * 4 + j] = yr;
      Gim[((size_t)f * kL + n) * 4 + j] = yi;
    }
    y4r = y3r; y4i = y3i; y3r = y2r; y3i = y2i;
    y2r = y1r; y2i = y1i; y1r = yr;  y1i = yi;
  }
}

// ---------------------------------------------------------------------------
// K1: zero-state chunk-end states. For chunk c, batch b, r=0..3:
//   S0[f][c*B+b][r] = sum_{m=0..L-1-r} h[m] * x_c[L-1-r-m]   (complex)
// ---------------------------------------------------------------------------
__global__ void gtfb_states_k(const float* __restrict__ x,
                              const float* __restrict__ Hre,
                              const float* __restrict__ Him,
                              float* __restrict__ S0re, float* __restrict__ S0im) {
  const int c = blockIdx.x, f = blockIdx.y;
  __shared__ float xsh[kB * kL];
  for (int i = threadIdx.x; i < kB * kL; i += 32)
    xsh[i] = x[(size_t)(i >> 8) * kT + c * kL + (i & 255)];
  __syncthreads();
  const int b = threadIdx.x >> 2, r = threadIdx.x & 3;
  const int nend = kL - 1 - r;
  const float* hr = Hre + f * kL;
  const float* hi = Him + f * kL;
  const float* xb = xsh + b * kL;
  float ar = 0.f, ai = 0.f;
  for (int m = 0; m <= nend; ++m) {
    const float xv = xb[nend - m];
    ar += hr[m] * xv;
    ai += hi[m] * xv;
  }
  const size_t idx = ((size_t)f * kN + c * kB + b) * 4 + r;
  S0re[idx] = ar;
  S0im[idx] = ai;
}

// ---------------------------------------------------------------------------
// K2: serial scan over the 64 chunks per (f,b) channel.
//   Sinit(c+1)[i] = S0(c)[i] + sum_j M[i][j]*Sinit(c)[j],  M[i][j]=G_j[L-1-i]
// ---------------------------------------------------------------------------
__global__ void gtfb_scan_k(const float* __restrict__ Gre,
                            const float* __restrict__ Gim,
                            const float* __restrict__ S0re,
                            const float* __restrict__ S0im,
                            float* __restrict__ Sire, float* __restrict__ Siim) {
  const int id = blockIdx.x * blockDim.x + threadIdx.x;
  if (id >= kF * kB) return;
  const int f = id >> 3, b = id & 7;
  float Mre[4][4], Mim[4][4];
  for (int i = 0; i < 4; ++i)
    for (int jj = 0; jj < 4; ++jj) {
      const size_t g = ((size_t)f * kL + (kL - 1 - i)) * 4 + jj;
      Mre[i][jj] = Gre[g];
      Mim[i][jj] = Gim[g];
    }
  float sr[4] = {0.f, 0.f, 0.f, 0.f}, si[4] = {0.f, 0.f, 0.f, 0.f};
  for (int c = 0; c < kC; ++c) {
    const size_t base = ((size_t)f * kN + c * kB + b) * 4;
    for (int jj = 0; jj < 4; ++jj) { Sire[base + jj] = sr[jj]; Siim[base + jj] = si[jj]; }
    float nr[4], ni[4];
    for (int i = 0; i < 4; ++i) {
      float tr = S0re[base + i], ti = S0im[base + i];
      for (int jj = 0; jj < 4; ++jj) {
        tr += Mre[i][jj] * sr[jj] - Mim[i][jj] * si[jj];
        ti += Mre[i][jj] * si[jj] + Mim[i][jj] * sr[jj];
      }
      nr[i] = tr; ni[i] = ti;
    }
    for (int i = 0; i < 4; ++i) { sr[i] = nr[i]; si[i] = ni[i]; }
  }
}

// ---------------------------------------------------------------------------
// K3: main WMMA kernel. For one (chunk c, batch b, 32-filter group):
//   out[t][f] = sum_k A[t][k] * HreT[k][f]  +  Re( G_f[t][:] . Sinit_f[:] )
// A[t][k] = x_c[t-k] for t>=k else 0 (zero-padded LDS stage).
// LDS operands staged by the Tensor Data Mover (wave 0 issues, TENSORcnt wait).
// ---------------------------------------------------------------------------
__global__ void __launch_bounds__(256, 1) gtfb_conv_wmma_k(
    const float* __restrict__ x, const float* __restrict__ HreT,
    const float* __restrict__ Gre, const float* __restrict__ Gim,
    const float* __restrict__ Sire, const float* __restrict__ Siim,
    float* __restrict__ out) {
  const int fbase = blockIdx.x * 32;
  const int c = blockIdx.y;
  const int b = blockIdx.z;
  // smem[0..255]   : zero causality pad      (xsh lower half)
  // smem[256..511] : x chunk (TDM tile)      (xsh upper half)
  // smem[512..]    : HreT slab [256 k][32 f] (TDM tile)
  __shared__ float smem[2 * kL + kL * 32];
  float* xsh = smem;
  float* hsh = smem + 2 * kL;
  const int tid = threadIdx.x;
  const int w = tid >> 5;
  const int lane = tid & 31;

  // Zero the causality pad (blockDim==256, tid 0..255 each zero one element).
  if (tid < kL) xsh[tid] = 0.f;

  if (w == 0) {
    const unsigned int lds_base = (unsigned int)(uintptr_t)smem;  // LDS aperture: low 32 bits
    // x chunk: 1D tile of 256 f32 -> smem[256..511]
    tdm_load_2d_f32(lds_base + kL * 4, x + (size_t)b * kT + (size_t)c * kL,
                    /*dim0=*/kL, /*dim1=*/1, /*stride0=*/kL,
                    /*tile0=*/kL, /*tile1=*/1);
    // HreT slab: 2D tile [256 rows x 32 cols] from [L][F] tensor -> smem[512..]
    tdm_load_2d_f32(lds_base + 2 * kL * 4, HreT + fbase,
                    /*dim0=*/kF, /*dim1=*/kL, /*stride0=*/kF,
                    /*tile0=*/32, /*tile1=*/kL);
    __builtin_amdgcn_s_wait_tensorcnt(0);
  }
  __syncthreads();

  const int mrow = lane & 15;
  const int khalf = lane >> 4;      // 0: K pair {k0,k0+1} / row k0 ; 1: {k0+2,k0+3} / row k0+1

  v8f acc[2][2] = {};
  const int tb0 = (w * 2) * 16;
  const int tb1 = (w * 2 + 1) * 16;

  for (int k0 = 0; k0 < kL; k0 += 4) {
    // A fragments (16x4: lanes 0-15 hold K={k0,k0+1}, lanes 16-31 hold {k0+2,k0+3})
    const int ka = k0 + khalf * 2;
    v2f afr0, afr1;
    afr0[0] = xsh[kL + tb0 + mrow - ka];
    afr0[1] = xsh[kL + tb0 + mrow - ka - 1];
    afr1[0] = xsh[kL + tb1 + mrow - ka];
    afr1[1] = xsh[kL + tb1 + mrow - ka - 1];
    // B fragments (4x16: VGPR0 rows {k0,k0+1} across half-waves, VGPR1 rows {k0+2,k0+3})
    const int kb = k0 + khalf;
    v2f bfr0, bfr1;
    bfr0[0] = hsh[kb * 32 + mrow];
    bfr0[1] = hsh[(kb + 2) * 32 + mrow];
    bfr1[0] = hsh[kb * 32 + 16 + mrow];
    bfr1[1] = hsh[(kb + 2) * 32 + 16 + mrow];

    acc[0][0] = __builtin_amdgcn_wmma_f32_16x16x4_f32(false, afr0, false, bfr0, (short)0, acc[0][0], false, false);
    acc[0][1] = __builtin_amdgcn_wmma_f32_16x16x4_f32(false, afr0, false, bfr1, (short)0, acc[0][1], false, false);
    acc[1][0] = __builtin_amdgcn_wmma_f32_16x16x4_f32(false, afr1, false, bfr0, (short)0, acc[1][0], false, false);
    acc[1][1] = __builtin_amdgcn_wmma_f32_16x16x4_f32(false, afr1, false, bfr1, (short)0, acc[1][1], false, false);
  }

  // Epilogue: add homogeneous correction Re(G . Sinit) and store (f-contiguous).
  const int nidx = c * kB + b;
  for (int fi = 0; fi < 2; ++fi) {
    const int fcol = fbase + fi * 16 + mrow;
    const float4 sre = *(const float4*)(Sire + ((size_t)fcol * kN + nidx) * 4);
    const float4 sim = *(const float4*)(Siim + ((size_t)fcol * kN + nidx) * 4);
    for (int ti = 0; ti < 2; ++ti) {
      const int tb = (ti == 0) ? tb0 : tb1;
#pragma unroll
      for (int r = 0; r < 8; ++r) {
        const int tl = tb + r + khalf * 8;   // C/D layout: VGPR r -> M=r (lanes<16) / M=r+8
        const float4 gre = *(const float4*)(Gre + ((size_t)fcol * kL + tl) * 4);
        const float4 gim = *(const float4*)(Gim + ((size_t)fcol * kL + tl) * 4);
        const float corr = gre.x * sre.x + gre.y * sre.y + gre.z * sre.z + gre.w * sre.w
                         - gim.x * sim.x - gim.y * sim.y - gim.z * sim.z - gim.w * sim.w;
        out[(size_t)b * (kT * kF) + (size_t)(c * kL + tl) * kF + fcol] = acc[ti][fi][r] + corr;
      }
    }
  }
}

// ---------------------------------------------------------------------------
extern "C" void kernel_launch(void* const* d_in, const int* in_sizes, int n_in,
                              void* d_out, int out_size, void* d_ws, size_t ws_size,
                              hipStream_t stream) {
  (void)in_sizes; (void)n_in; (void)out_size; (void)ws_size;
  const float* x  = (const float*)d_in[0];
  const float* nf = (const float*)d_in[1];
  const float* a1 = (const float*)d_in[2];   // complex64 -> interleaved float pairs
  const float* a2 = (const float*)d_in[3];
  const float* a3 = (const float*)d_in[4];
  const float* a4 = (const float*)d_in[5];
  float* out = (float*)d_out;

  float* ws   = (float*)d_ws;                // ~11.3 MB total
  float* Hre  = ws;                          // [F][L]
  float* Him  = Hre  + kF * kL;              // [F][L]
  float* HreT = Him  + kF * kL;              // [L][F]
  float* Gre  = HreT + kF * kL;              // [F][L][4]
  float* Gim  = Gre  + (size_t)kF * kL * 4;  // [F][L][4]
  float* S0re = Gim  + (size_t)kF * kL * 4;  // [F][N][4]
  float* S0im = S0re + (size_t)kF * kN * 4;
  float* Sire = S0im + (size_t)kF * kN * 4;  // [F][N][4]
  float* Siim = Sire + (size_t)kF * kN * 4;

  gtfb_coeffs_k<<<dim3(kF), dim3(8), 0, stream>>>(nf, a1, a2, a3, a4,
                                                  Hre, Him, HreT, Gre, Gim);
  gtfb_states_k<<<dim3(kC, kF), dim3(32), 0, stream>>>(x, Hre, Him, S0re, S0im);
  gtfb_scan_k<<<dim3((kF * kB + 255) / 256), dim3(256), 0, stream>>>(Gre, Gim,
                                                                     S0re, S0im,
                                                                     Sire, Siim);
  gtfb_conv_wmma_k<<<dim3(kF / 32, kC, kB), dim3(256), 0, stream>>>(x, HreT, Gre, Gim,
                                                                    Sire, Siim, out);
}